// Gr4j4Dpl_3418793967625
// MI455X (gfx1250) — compile-verified
//
#include <hip/hip_runtime.h>
#include <hip/hip_bf16.h>

#define AS1 __attribute__((address_space(1)))
#define AS3 __attribute__((address_space(3)))

typedef int v2i __attribute__((ext_vector_type(2)));

constexpr int THREADS = 256;   // 8 wave32s per block
constexpr int CHUNK   = 16;    // timesteps staged per async chunk

// ---------------- fast math (CDNA5 hardware transcendentals) ----------------
__device__ __forceinline__ float fast_rcp(float x) {
#if __has_builtin(__builtin_amdgcn_rcpf)
  return __builtin_amdgcn_rcpf(x);
#else
  return 1.0f / x;
#endif
}
__device__ __forceinline__ float fast_rsq(float x) {
#if __has_builtin(__builtin_amdgcn_rsqf)
  return __builtin_amdgcn_rsqf(x);
#else
  return rsqrtf(x);
#endif
}
__device__ __forceinline__ float fast_sqrt(float x) {
#if __has_builtin(__builtin_amdgcn_sqrtf)
  return __builtin_amdgcn_sqrtf(x);
#else
  return sqrtf(x);
#endif
}
// tanh for |x| <= ~0.02 (guaranteed here: arg = p/x1, p<1, x1>=100).
// Hardware v_tanh_f32 if available; else 5th-order odd series, fp32-exact in range.
__device__ __forceinline__ float fast_tanh_small(float x) {
#if __has_builtin(__builtin_amdgcn_tanhf)
  return __builtin_amdgcn_tanhf(x);            // v_tanh_f32 on gfx1250
#elif __has_builtin(__builtin_amdgcn_tanh_f32)
  return __builtin_amdgcn_tanh_f32(x);
#else
  const float x2 = x * x;
  return x * (1.0f - 0.33333333f * x2 * (1.0f - 0.4f * x2));
#endif
}
// x^2.5 for x >= 0
__device__ __forceinline__ float pow25(float x) { return x * x * fast_sqrt(x); }

// ---------------- gfx1250 async global->LDS copy ----------------
__device__ __forceinline__ void async_copy_b64(const void* g, void* l) {
#if __has_builtin(__builtin_amdgcn_global_load_async_to_lds_b64)
  // builtin is typed by payload: b64 -> pointers to <2 x i32>
  __builtin_amdgcn_global_load_async_to_lds_b64(
      (AS1 v2i*)(unsigned long long)g,
      (AS3 v2i*)(unsigned)(unsigned long long)l,
      0, 0);
#else
  unsigned lds_off = (unsigned)(unsigned long long)l;  // flat LDS addr low bits = LDS offset
  asm volatile("global_load_async_to_lds_b64 %0, %1, off"
               :: "v"(lds_off), "v"(g) : "memory");
#endif
}

template <int N>
__device__ __forceinline__ void wait_asynccnt() {
#if __has_builtin(__builtin_amdgcn_s_wait_asynccnt)
  __builtin_amdgcn_s_wait_asynccnt(N);
  asm volatile("" ::: "memory");
#else
  asm volatile("s_wait_asynccnt %0" :: "i"(N) : "memory");
#endif
}

// Issue CHUNK async B64 loads: this lane's (p,e) pair for timesteps t0..t0+CHUNK-1.
__device__ __forceinline__ void issue_chunk(const float2* __restrict__ pe2,
                                            int t0, int T, int B, int b,
                                            float2* dst0 /* &sbuf[bi][0][tid] */) {
  // make sure prior DS reads of this buffer have retired before async engine overwrites it
  asm volatile("s_wait_dscnt 0x0" ::: "memory");
#pragma unroll
  for (int k = 0; k < CHUNK; ++k) {
    int t = t0 + k;
    t = (t < T) ? t : (T - 1);   // clamp: keeps count uniform; clamped rows never consumed
    async_copy_b64((const void*)(pe2 + ((size_t)t * B + b)),
                   (void*)(dst0 + (size_t)k * THREADS));
  }
}

// ---------------- fused GR4J: production scan + UH conv + routing scan ----------------
__global__ __launch_bounds__(THREADS, 1)
void gr4j_fused_kernel(const float* __restrict__ pe,      // [T, B, 2]
                       const float* __restrict__ params,  // [B, 4]
                       float* __restrict__ out,           // [T - warmup, B]
                       int T, int B, int warmup) {
  __shared__ float2 sbuf[2][CHUNK][THREADS];   // 64 KB, double-buffered staging

  const int tid = threadIdx.x;
  const int b   = blockIdx.x * THREADS + tid;
  if (b >= B) return;

  // ---- parameters ----
  const float4 pp = ((const float4*)params)[b];
  const float x1 = 100.0f + pp.x * 1100.0f;
  const float x2 =  -5.0f + pp.y *    8.0f;
  const float x3 =  20.0f + pp.z *  280.0f;
  const float x4 =   1.1f + pp.w *    1.8f;
  const float inv_x1 = fast_rcp(x1);
  const float inv_x3 = fast_rcp(x3);
  const float inv_x4 = fast_rcp(x4);

  // ---- unit hydrograph kernels (K1=3, K2=6) ----
  float sh1[4];
#pragma unroll
  for (int k = 0; k <= 3; ++k)
    sh1[k] = pow25(fminf((float)k * inv_x4, 1.0f));
  const float uh1_0 = sh1[1] - sh1[0];
  const float uh1_1 = sh1[2] - sh1[1];
  const float uh1_2 = sh1[3] - sh1[2];

  float sh2[7];
#pragma unroll
  for (int k = 0; k <= 6; ++k) {
    const float tf = (float)k;
    if (tf <= x4) sh2[k] = 0.5f * pow25(tf * inv_x4);
    else          sh2[k] = 1.0f - 0.5f * pow25(fmaxf(2.0f - tf * inv_x4, 0.0f));
  }
  const float uh2_0 = sh2[1] - sh2[0];
  const float uh2_1 = sh2[2] - sh2[1];
  const float uh2_2 = sh2[3] - sh2[2];
  const float uh2_3 = sh2[4] - sh2[3];
  const float uh2_4 = sh2[5] - sh2[4];
  const float uh2_5 = sh2[6] - sh2[5];

  // ---- state ----
  float s = 0.5f * x1;               // production store (warmup init)
  float r = 0.5f * x3;               // routing store (warmup init)
  float h0 = 0.f, h1 = 0.f, h2 = 0.f, h3 = 0.f, h4 = 0.f, h5 = 0.f;  // prs history

  const float2* pe2 = (const float2*)pe;    // element (t*B + b) = {p, e}
  const int nchunk = (T + CHUNK - 1) / CHUNK;

  issue_chunk(pe2, 0, T, B, b, &sbuf[0][0][tid]);

  int t = 0;
  for (int c = 0; c < nchunk; ++c) {
    const int bi = c & 1;
    if (c + 1 < nchunk) {
      issue_chunk(pe2, (c + 1) * CHUNK, T, B, b, &sbuf[bi ^ 1][0][tid]);
      wait_asynccnt<CHUNK>();   // chunk c retired (async loads complete in order)
    } else {
      wait_asynccnt<0>();
    }

    // Constant trip count: steps with t >= T compute dead state (never stored, never
    // read back) from the clamped duplicate row -- safe, and lets the unroller overlap
    // the state-independent work (ds_load, pn/en, tanh) with the previous step's chain.
#pragma unroll 4
    for (int k = 0; k < CHUNK; ++k, ++t) {
      const float2 v = sbuf[bi][k][tid];
      const float p  = v.x, e = v.y;
      const float dn = p - e;
      const float pn = fmaxf(dn, 0.0f);
      const float en = fmaxf(-dn, 0.0f);

      // ---- production store ----
      const float tp = fast_tanh_small(pn * inv_x1);   // state-independent
      const float te = fast_tanh_small(en * inv_x1);   // state-independent
      const float sc = fminf(fmaxf(s, 0.0f), x1);
      const float sr = sc * inv_x1;
      const float ps = x1 * (1.0f - sr * sr) * tp * fast_rcp(1.0f + sr * tp);
      const float es = sc * (2.0f - sr) * te * fast_rcp(1.0f + (1.0f - sr) * te);
      const float s2 = fminf(fmaxf(sc - es + ps, 0.0f), x1);
      const float z  = 0.44444445f * s2 * inv_x1;
      const float z4 = (z * z) * (z * z);
      const float perc = s2 * (1.0f - fast_rsq(fast_sqrt(1.0f + z4)));  // (1+z^4)^(-1/4)
      s = s2 - perc;
      const float pr = perc + (pn - ps);

      // ---- causal FIR (ring in registers) ----
      h5 = h4; h4 = h3; h3 = h2; h2 = h1; h1 = h0; h0 = pr;
      const float q9 = fmaf(uh1_0, h0, fmaf(uh1_1, h1, uh1_2 * h2));
      const float q1 = fmaf(uh2_0, h0, fmaf(uh2_1, h1,
                       fmaf(uh2_2, h2, fmaf(uh2_3, h3,
                       fmaf(uh2_4, h4, uh2_5 * h5)))));

      // ---- routing store ----
      const float rc  = fminf(fmaxf(r, 0.0f), x3);
      const float rr  = rc * inv_x3;
      const float gex = x2 * (rr * rr * rr) * fast_sqrt(rr);   // rr^3.5
      const float r2  = fmaxf(0.0f, rc + q9 + gex);
      const float w   = r2 * inv_x3;
      const float w4  = (w * w) * (w * w);
      const float qr  = r2 * (1.0f - fast_rsq(fast_sqrt(1.0f + w4)));
      r = r2 - qr;
      const float qd  = fmaxf(0.0f, q1 + gex);

      if (t >= warmup && t < T) out[(size_t)(t - warmup) * B + b] = qr + qd;
      if (t == warmup - 1) { h0 = h1 = h2 = h3 = h4 = h5 = 0.0f; }  // conv restarts at main segment
    }
  }
}

extern "C" void kernel_launch(void* const* d_in, const int* in_sizes, int n_in,
                              void* d_out, int out_size, void* d_ws, size_t ws_size,
                              hipStream_t stream) {
  (void)n_in; (void)d_ws; (void)ws_size; (void)out_size;
  const float* pe     = (const float*)d_in[0];   // [T, B, 2] f32
  const float* params = (const float*)d_in[1];   // [B, 4]    f32
  float* out          = (float*)d_out;           // [T-365, B, 1] f32

  const int B = in_sizes[1] / 4;
  const int T = in_sizes[0] / (2 * B);
  const int warmup = 365;

  dim3 block(THREADS);
  dim3 grid((B + THREADS - 1) / THREADS);
  gr4j_fused_kernel<<<grid, block, 0, stream>>>(pe, params, out, T, B, warmup);
}